// TransformerConv_22514218566261
// MI455X (gfx1250) — compile-verified
//
#include <hip/hip_runtime.h>
#include <cmath>

#define NN 50000
#define NE 600000

typedef __attribute__((ext_vector_type(2))) float v2f;
typedef __attribute__((ext_vector_type(8))) float v8f;

// ---------------------------------------------------------------------------
// LayerNorm over 64 channels, one wave32 per row, 2 elements per lane.
// ---------------------------------------------------------------------------
__global__ void ln64_kernel(const float* __restrict__ x,
                            const float* __restrict__ g,
                            const float* __restrict__ b,
                            float* __restrict__ out, int nrows) {
    int wid  = (blockIdx.x * blockDim.x + threadIdx.x) >> 5;
    int lane = threadIdx.x & 31;
    if (wid >= nrows) return;
    const float* xr = x + (size_t)wid * 64;
    float x0 = xr[lane], x1 = xr[lane + 32];
    float s = x0 + x1;
    for (int o = 16; o > 0; o >>= 1) s += __shfl_xor(s, o, 32);
    float mean = s * (1.0f / 64.0f);
    float d0 = x0 - mean, d1 = x1 - mean;
    float vs = d0 * d0 + d1 * d1;
    for (int o = 16; o > 0; o >>= 1) vs += __shfl_xor(vs, o, 32);
    float inv = rsqrtf(vs * (1.0f / 64.0f) + 1e-5f);
    float* orow = out + (size_t)wid * 64;
    orow[lane]      = d0 * inv * g[lane]      + b[lane];
    orow[lane + 32] = d1 * inv * g[lane + 32] + b[lane + 32];
}

// ---------------------------------------------------------------------------
// Generic fp32 GEMM via V_WMMA_F32_16X16X4_F32. One wave per 16x16 tile.
// D = act(A@B + bias) + resid.  All dims multiples of 16 (K multiple of 4).
// ---------------------------------------------------------------------------
__global__ void gemm_wmma_kernel(const float* __restrict__ A, int lda,
                                 const float* __restrict__ B, int ldb,
                                 const float* __restrict__ bias,
                                 const float* __restrict__ resid,
                                 float* __restrict__ D, int ldd,
                                 int K, int act) {
    int n0 = blockIdx.x * 16;
    int m0 = blockIdx.y * 16;
    int lane = threadIdx.x & 31;
    int half = lane >> 4;      // 0: K pair {k0,k0+1}, 1: {k0+2,k0+3}
    int lr   = lane & 15;
    const float* arow = A + (size_t)(m0 + lr) * lda;   // A row per lane
    const float* bcol = B + (n0 + lr);                 // B column per lane
    v8f c = {};
    for (int k0 = 0; k0 < K; k0 += 4) {
        v2f a, b;
        a.x = arow[k0 + 2 * half];
        a.y = arow[k0 + 2 * half + 1];
        b.x = bcol[(size_t)(k0 + 2 * half) * ldb];
        b.y = bcol[(size_t)(k0 + 2 * half + 1) * ldb];
        c = __builtin_amdgcn_wmma_f32_16x16x4_f32(false, a, false, b,
                                                  (short)0, c, false, false);
    }
#pragma unroll
    for (int r = 0; r < 8; ++r) {
        int row = m0 + half * 8 + r;
        int col = n0 + lr;
        float val = c[r];
        if (bias)  val += bias[col];
        if (act)   val = 0.5f * val * (1.0f + erff(val * 0.70710678118654752f));
        if (resid) val += resid[(size_t)row * ldd + col];
        D[(size_t)row * ldd + col] = val;
    }
}

// ---------------------------------------------------------------------------
// Edge attention phase
// ---------------------------------------------------------------------------
__device__ inline float atomicMaxF(float* addr, float value) {
    if (value >= 0.0f)
        return __int_as_float(atomicMax((int*)addr, __float_as_int(value)));
    return __uint_as_float(atomicMin((unsigned int*)addr, __float_as_uint(value)));
}

__global__ void init_kernel(float* __restrict__ amax, float* __restrict__ den,
                            float* __restrict__ agg) {
    long long i = blockIdx.x * (long long)blockDim.x + threadIdx.x;
    if (i < (long long)NN * 128) agg[i] = 0.0f;
    if (i < (long long)NN * 2) { amax[i] = -INFINITY; den[i] = 0.0f; }
}

// alpha[e,h] = <q[tgt,h,:], k[src,h,:] + e[e,h,:]> / 8 ; amax = segmax(alpha)
__global__ void edge_alpha_kernel(const int* __restrict__ ei,
                                  const float* __restrict__ q,
                                  const float* __restrict__ k,
                                  const float* __restrict__ e,
                                  float* __restrict__ alpha,
                                  float* __restrict__ amax) {
    int eid  = (blockIdx.x * blockDim.x + threadIdx.x) >> 5;
    int lane = threadIdx.x & 31;
    if (eid >= NE) return;
    int src = ei[eid], tgt = ei[NE + eid];
    const float* qr = q + (size_t)tgt * 128;
    const float* kr = k + (size_t)src * 128;
    const float* er = e + (size_t)eid * 128;
    float p0 = qr[lane]      * (kr[lane]      + er[lane])
             + qr[lane + 32] * (kr[lane + 32] + er[lane + 32]);
    float p1 = qr[lane + 64] * (kr[lane + 64] + er[lane + 64])
             + qr[lane + 96] * (kr[lane + 96] + er[lane + 96]);
    for (int o = 16; o > 0; o >>= 1) {
        p0 += __shfl_xor(p0, o, 32);
        p1 += __shfl_xor(p1, o, 32);
    }
    if (lane == 0) {
        float a0 = p0 * 0.125f, a1 = p1 * 0.125f;   // 1/sqrt(64)
        alpha[(size_t)eid * 2]     = a0;
        alpha[(size_t)eid * 2 + 1] = a1;
        atomicMaxF(&amax[(size_t)tgt * 2],     a0);
        atomicMaxF(&amax[(size_t)tgt * 2 + 1], a1);
    }
}

// ex = exp(alpha - amax[tgt]); den = segsum(ex); alpha buffer <- ex
__global__ void edge_ex_kernel(const int* __restrict__ ei,
                               float* __restrict__ alpha,
                               const float* __restrict__ amax,
                               float* __restrict__ den) {
    long long i = blockIdx.x * (long long)blockDim.x + threadIdx.x;
    if (i >= (long long)NE * 2) return;
    int eid = (int)(i >> 1), h = (int)(i & 1);
    int tgt = ei[NE + eid];
    float ex = expf(alpha[i] - amax[(size_t)tgt * 2 + h]);
    alpha[i] = ex;
    atomicAdd(&den[(size_t)tgt * 2 + h], ex);
}

// agg[tgt] += a * (v[src] + e[eid])
__global__ void edge_agg_kernel(const int* __restrict__ ei,
                                const float* __restrict__ alpha,
                                const float* __restrict__ den,
                                const float* __restrict__ v,
                                const float* __restrict__ e,
                                float* __restrict__ agg) {
    int eid  = (blockIdx.x * blockDim.x + threadIdx.x) >> 5;
    int lane = threadIdx.x & 31;
    if (eid >= NE) return;
    int src = ei[eid], tgt = ei[NE + eid];
    float a0 = alpha[(size_t)eid * 2]     / (den[(size_t)tgt * 2]     + 1e-8f);
    float a1 = alpha[(size_t)eid * 2 + 1] / (den[(size_t)tgt * 2 + 1] + 1e-8f);
    const float* vr = v + (size_t)src * 128;
    const float* er = e + (size_t)eid * 128;
    float* ar = agg + (size_t)tgt * 128;
    atomicAdd(&ar[lane],      a0 * (vr[lane]      + er[lane]));
    atomicAdd(&ar[lane + 32], a0 * (vr[lane + 32] + er[lane + 32]));
    atomicAdd(&ar[lane + 64], a1 * (vr[lane + 64] + er[lane + 64]));
    atomicAdd(&ar[lane + 96], a1 * (vr[lane + 96] + er[lane + 96]));
}

// ---------------------------------------------------------------------------
extern "C" void kernel_launch(void* const* d_in, const int* in_sizes, int n_in,
                              void* d_out, int out_size, void* d_ws, size_t ws_size,
                              hipStream_t stream) {
    (void)in_sizes; (void)n_in; (void)out_size; (void)ws_size;
    const float* x    = (const float*)d_in[0];
    const int*   ei   = (const int*)d_in[1];     // (2,E): row 0 = src, row 1 = tgt
    const float* ea   = (const float*)d_in[2];
    const float* Wq   = (const float*)d_in[3];
    const float* bq   = (const float*)d_in[4];
    const float* Wk   = (const float*)d_in[5];
    const float* bk   = (const float*)d_in[6];
    const float* Wv   = (const float*)d_in[7];
    const float* bv   = (const float*)d_in[8];
    const float* We   = (const float*)d_in[9];
    const float* Wp   = (const float*)d_in[10];
    const float* bp   = (const float*)d_in[11];
    const float* ln1g = (const float*)d_in[12];
    const float* ln1b = (const float*)d_in[13];
    const float* ln2g = (const float*)d_in[14];
    const float* ln2b = (const float*)d_in[15];
    const float* W1   = (const float*)d_in[16];
    const float* b1   = (const float*)d_in[17];
    const float* W2   = (const float*)d_in[18];
    const float* b2   = (const float*)d_in[19];
    float* out = (float*)d_out;

    float* ws = (float*)d_ws;
    size_t off = 0;
    float* h   = ws + off; off += (size_t)NN * 64;
    float* q   = ws + off; off += (size_t)NN * 128;
    float* kb  = ws + off; off += (size_t)NN * 128;
    float* vb  = ws + off; off += (size_t)NN * 128;
    float* e   = ws + off; off += (size_t)NE * 128;
    float* alp = ws + off; off += (size_t)NE * 2;
    float* amx = ws + off; off += (size_t)NN * 2;
    float* den = ws + off; off += (size_t)NN * 2;
    float* agg = ws + off; off += (size_t)NN * 128;
    float* o1  = ws + off; off += (size_t)NN * 64;
    float* h2  = ws + off; off += (size_t)NN * 64;
    float* f1  = ws + off; off += (size_t)NN * 64;

    // 1. LN1
    ln64_kernel<<<(NN + 7) / 8, 256, 0, stream>>>(x, ln1g, ln1b, h, NN);

    // 2. Q/K/V projections: (N,64) @ (64,128) + bias
    dim3 gqkv(128 / 16, NN / 16);
    gemm_wmma_kernel<<<gqkv, 32, 0, stream>>>(h, 64, Wq, 128, bq, nullptr, q,  128, 64, 0);
    gemm_wmma_kernel<<<gqkv, 32, 0, stream>>>(h, 64, Wk, 128, bk, nullptr, kb, 128, 64, 0);
    gemm_wmma_kernel<<<gqkv, 32, 0, stream>>>(h, 64, Wv, 128, bv, nullptr, vb, 128, 64, 0);

    // 3. Edge projection: (E,32) @ (32,128)
    dim3 ge(128 / 16, NE / 16);
    gemm_wmma_kernel<<<ge, 32, 0, stream>>>(ea, 32, We, 128, nullptr, nullptr, e, 128, 32, 0);

    // 4. init amax/-inf, den/agg zero (must reset every call: atomics accumulate)
    init_kernel<<<((long long)NN * 128 + 255) / 256, 256, 0, stream>>>(amx, den, agg);

    // 5. attention scores + segment max
    edge_alpha_kernel<<<NE / 8, 256, 0, stream>>>(ei, q, kb, e, alp, amx);
    // 6. exp + segment sum
    edge_ex_kernel<<<((long long)NE * 2 + 255) / 256, 256, 0, stream>>>(ei, alp, amx, den);
    // 7. weighted aggregation
    edge_agg_kernel<<<NE / 8, 256, 0, stream>>>(ei, alp, den, vb, e, agg);

    // 8. output projection (N,128)@(128,64)+bp + residual x
    dim3 gp(64 / 16, NN / 16);
    gemm_wmma_kernel<<<gp, 32, 0, stream>>>(agg, 128, Wp, 64, bp, x, o1, 64, 128, 0);

    // 9. LN2
    ln64_kernel<<<(NN + 7) / 8, 256, 0, stream>>>(o1, ln2g, ln2b, h2, NN);

    // 10. FFN: gelu(h2@W1+b1) @ W2 + b2 + o1
    gemm_wmma_kernel<<<gp, 32, 0, stream>>>(h2, 64, W1, 64, b1, nullptr, f1, 64, 64, 1);
    gemm_wmma_kernel<<<gp, 32, 0, stream>>>(f1, 64, W2, 64, b2, o1, out, 64, 64, 0);
}